// Net_45741401703047
// MI455X (gfx1250) — compile-verified
//
#include <hip/hip_runtime.h>

typedef __attribute__((ext_vector_type(2))) float v2f;
typedef __attribute__((ext_vector_type(8))) float v8f;

#define D 12

// ---------------------------------------------------------------------------
// Kernel 1: zero the aggregation workspace (agg[N*D] followed by cnt[N]).
// ---------------------------------------------------------------------------
__global__ void zero_ws(float* __restrict__ p, int n) {
    int i = blockIdx.x * blockDim.x + threadIdx.x;
    if (i < n) p[i] = 0.0f;
}

// ---------------------------------------------------------------------------
// Kernel 2: edge-parallel gather + scatter-add.
// edge_index (102 MB) is streamed once -> non-temporal loads keep it from
// evicting the L2-resident x (4.8 MB) and agg (4.8 MB) working set.
// Atomics resolve at L2 (default RT policy, non-returning).
// ---------------------------------------------------------------------------
__global__ void edge_scatter(const float* __restrict__ x,
                             const long long* __restrict__ ei,
                             float* __restrict__ agg,
                             float* __restrict__ cnt,
                             int E) {
    int e = blockIdx.x * blockDim.x + threadIdx.x;
    if (e >= E) return;
    long long s = __builtin_nontemporal_load(ei + e);                 // src
    long long d = __builtin_nontemporal_load(ei + (long long)E + e);  // dst
    const float4* xs = (const float4*)(x + s * D);   // rows are 48B => 16B aligned
    float4 a0 = xs[0], a1 = xs[1], a2 = xs[2];
    float* ad = agg + d * D;
    atomicAdd(ad + 0,  a0.x); atomicAdd(ad + 1,  a0.y);
    atomicAdd(ad + 2,  a0.z); atomicAdd(ad + 3,  a0.w);
    atomicAdd(ad + 4,  a1.x); atomicAdd(ad + 5,  a1.y);
    atomicAdd(ad + 6,  a1.z); atomicAdd(ad + 7,  a1.w);
    atomicAdd(ad + 8,  a2.x); atomicAdd(ad + 9,  a2.y);
    atomicAdd(ad + 10, a2.z); atomicAdd(ad + 11, a2.w);
    atomicAdd(cnt + d, 1.0f);
}

// ---------------------------------------------------------------------------
// Kernel 3: out = mean @ Wl^T + x @ Wr^T + b via V_WMMA_F32_16X16X4_F32.
// One wave32 owns a 16-node x 16-col tile. K=12 handled by 3 WMMA steps per
// GEMM (lane-half carries k-pairs {4s,4s+1} / {4s+2,4s+3}); no K padding.
// A-layout: lane m (=lane&15) holds row m, vgpr0/1 = k0/k0+1 (ISA 7.12.2).
// B-layout: lane n holds column n of B = W^T, i.e. B[k][n] = W[n][k].
// C-layout: vgpr v, lane-half h -> row v + 8h, col lane&15.
// Output is written once and never re-read -> non-temporal stores.
// ---------------------------------------------------------------------------
__global__ void sage_wmma(const float* __restrict__ x,
                          const float* __restrict__ Wl,
                          const float* __restrict__ Wr,
                          const float* __restrict__ bias,
                          const float* __restrict__ agg,
                          const float* __restrict__ cnt,
                          float* __restrict__ out,
                          int ntiles) {
    int lane = threadIdx.x & 31;
    int wave = threadIdx.x >> 5;
    int tile = blockIdx.x * (blockDim.x >> 5) + wave;
    if (tile >= ntiles) return;          // uniform per wave: EXEC stays all-1s

    int half = lane >> 4;                // 0: k offsets {0,1}; 1: {2,3}
    int m    = lane & 15;                // A row within tile / B column
    long long row = (long long)tile * 16 + m;   // N is a multiple of 16

    float inv = 1.0f / fmaxf(cnt[row], 1.0f);
    const float* ar = agg + row * D;
    const float* xr = x   + row * D;

    bool ncol = (m < D);
    int  nc   = ncol ? m : 0;            // clamp so loads are always in-bounds

    v8f acc = {};

    // GEMM 1: mean @ Wl^T
    #pragma unroll
    for (int s = 0; s < 3; ++s) {
        int k0 = 4 * s + 2 * half;       // k0, k0+1 in [0,12)
        v2f a, b;
        a.x = ar[k0]     * inv;
        a.y = ar[k0 + 1] * inv;
        float w0 = Wl[nc * D + k0];
        float w1 = Wl[nc * D + k0 + 1];
        b.x = ncol ? w0 : 0.0f;          // v_cndmask, no EXEC change
        b.y = ncol ? w1 : 0.0f;
        acc = __builtin_amdgcn_wmma_f32_16x16x4_f32(
            false, a, false, b, (short)0, acc, false, false);
    }
    // GEMM 2: x @ Wr^T, accumulated into the same C
    #pragma unroll
    for (int s = 0; s < 3; ++s) {
        int k0 = 4 * s + 2 * half;
        v2f a, b;
        a.x = xr[k0];
        a.y = xr[k0 + 1];
        float w0 = Wr[nc * D + k0];
        float w1 = Wr[nc * D + k0 + 1];
        b.x = ncol ? w0 : 0.0f;
        b.y = ncol ? w1 : 0.0f;
        acc = __builtin_amdgcn_wmma_f32_16x16x4_f32(
            false, a, false, b, (short)0, acc, false, false);
    }

    if (ncol) {                          // columns 12..15 are padding
        float bv = bias[m];
        #pragma unroll
        for (int v = 0; v < 8; ++v) {
            long long orow = (long long)tile * 16 + v + 8 * half;
            __builtin_nontemporal_store(acc[v] + bv, out + orow * D + m);
        }
    }
}

// ---------------------------------------------------------------------------
extern "C" void kernel_launch(void* const* d_in, const int* in_sizes, int n_in,
                              void* d_out, int out_size, void* d_ws, size_t ws_size,
                              hipStream_t stream) {
    const float*     x   = (const float*)d_in[0];
    const float*     Wl  = (const float*)d_in[1];
    const float*     Wr  = (const float*)d_in[2];
    const float*     b   = (const float*)d_in[3];
    const long long* ei  = (const long long*)d_in[4];   // int64 edge_index [2,E]
    float*           out = (float*)d_out;

    int N = in_sizes[0] / D;   // 100000
    int E = in_sizes[4] / 2;   // 6400000

    float* agg = (float*)d_ws;              // N*D floats
    float* cnt = agg + (size_t)N * D;       // N floats  (total ~5.2 MB of ws)

    int tot = N * (D + 1);
    zero_ws<<<dim3((tot + 255) / 256), dim3(256), 0, stream>>>(agg, tot);

    edge_scatter<<<dim3((E + 255) / 256), dim3(256), 0, stream>>>(x, ei, agg, cnt, E);

    int ntiles = (N + 15) / 16;             // 6250, exact
    int wpb = 8;                            // 8 waves (256 threads) per block
    sage_wmma<<<dim3((ntiles + wpb - 1) / wpb), dim3(wpb * 32), 0, stream>>>(
        x, Wl, Wr, b, agg, cnt, out, ntiles);
}